// EncodingP_86526411145826
// MI455X (gfx1250) — compile-verified
//
#include <hip/hip_runtime.h>
#include <hip/hip_bf16.h>

// Problem constants from the reference
#define B_ 4
#define D_ 256
#define K_ 32
#define N_ 4096   // H*W = 64*64

typedef __attribute__((ext_vector_type(2))) float v2f;   // A/B frag of V_WMMA_F32_16X16X4_F32 (16x4 f32 = 2 VGPRs)
typedef __attribute__((ext_vector_type(8))) float v8f;   // 16x16 f32 C/D accumulator (8 VGPRs)

// -------------------------------------------------------------------------
// Zero-init for the global asum accumulator (graph-safe, deterministic).
// -------------------------------------------------------------------------
__global__ void enc_zero_kernel(float* __restrict__ p, int n) {
  int i = blockIdx.x * blockDim.x + threadIdx.x;
  if (i < n) p[i] = 0.0f;
}

// -------------------------------------------------------------------------
// Pass 1: per block = 128 pixels of one batch image.
//   xc tile via WMMA f32 16x16x4, x2 fused into the A-fragment loads,
//   softmax over K=32 per pixel, A written to workspace, asum accumulated.
// -------------------------------------------------------------------------
__global__ __launch_bounds__(256) void enc_assign_kernel(
    const float* __restrict__ X,       // (B, D, N)  d-major, n contiguous
    const float* __restrict__ Cw,      // (K, D)
    const float* __restrict__ Sc,      // (K,)
    float* __restrict__ wsA,           // (B, N, K)
    float* __restrict__ wsAsum)        // (B, K)
{
  __shared__ float sC[K_ * D_];        // 32 KB codewords
  __shared__ float sXC[128 * K_];      // 16 KB xc tile
  __shared__ float sX2[128];
  __shared__ float sC2[K_];
  __shared__ float sScale[K_];
  __shared__ float sAsum[K_];

  const int tid  = threadIdx.x;
  const int b    = blockIdx.x >> 5;          // 32 pixel-tiles per batch
  const int tile = blockIdx.x & 31;
  const int n0b  = tile * 128;

  // Stage codewords into LDS with 128-bit loads/stores
  {
    const float4* g4 = (const float4*)Cw;
    float4* s4 = (float4*)sC;
    for (int i = tid; i < (K_ * D_) / 4; i += 256) s4[i] = g4[i];
  }
  if (tid < K_) { sScale[tid] = Sc[tid]; sAsum[tid] = 0.0f; }
  __syncthreads();

  if (tid < K_) {                            // c2[k] = ||c_k||^2
    const float* row = &sC[tid * D_];
    float s = 0.0f;
    for (int d = 0; d < D_; ++d) s += row[d] * row[d];
    sC2[tid] = s;
  }
  __syncthreads();

  // ---- WMMA phase: wave w owns pixels [n0b + 16w, n0b + 16w + 16) ----
  const int wave = tid >> 5;
  const int lane = tid & 31;
  const int m    = lane & 15;                // pixel row within tile / B-col
  const int lh   = lane >> 4;                // lane half (K-split per ISA layout)
  const int n0   = n0b + wave * 16;
  const float* Xb = X + (size_t)b * D_ * N_;

  v8f acc0 = {};                             // codewords 0..15
  v8f acc1 = {};                             // codewords 16..31
  float sq = 0.0f;                           // partial ||x||^2 (this lane's d-subset)

  for (int d0 = 0; d0 < D_; d0 += 4) {
    const int da = d0 + 2 * lh;
    // A frag: (pixel m, d = da, da+1); d-stride is N floats in memory
    v2f a;
    a.x = Xb[(size_t)da * N_ + n0 + m];
    a.y = Xb[(size_t)(da + 1) * N_ + n0 + m];
    sq += a.x * a.x + a.y * a.y;
    // B frags from LDS: (d rows da, da+1; codeword col m) -> contiguous pair, ds_load_b64
    v2f b0 = *(const v2f*)&sC[(m) * D_ + da];
    v2f b1 = *(const v2f*)&sC[(16 + m) * D_ + da];
    acc0 = __builtin_amdgcn_wmma_f32_16x16x4_f32(false, a, false, b0, (short)0, acc0, false, false);
    acc1 = __builtin_amdgcn_wmma_f32_16x16x4_f32(false, a, false, b1, (short)0, acc1, false, false);
  }

  // Each lane-half saw complementary d's for pixel m: combine across halves (wave32)
  float sqTot = sq + __shfl_xor(sq, 16, 32);
  if (lane < 16) sX2[wave * 16 + m] = sqTot;

  // C/D layout: VGPR v, lane L -> (row v + 8*(L/16), col L%16)
  #pragma unroll
  for (int v = 0; v < 8; ++v) {
    const int p = wave * 16 + v + 8 * lh;
    sXC[p * K_ + m]      = acc0[v];
    sXC[p * K_ + 16 + m] = acc1[v];
  }
  __syncthreads();

  // ---- Softmax phase: one pixel per thread (threads 0..127) ----
  if (tid < 128) {
    const int p = tid;
    const int n = n0b + p;
    const float x2 = sX2[p];
    float t[K_];
    float mx = -3.4e38f;
    #pragma unroll
    for (int k = 0; k < K_; ++k) {
      const float L = x2 - 2.0f * sXC[p * K_ + k] + sC2[k];
      t[k] = L * sScale[k];
      mx = fmaxf(mx, t[k]);
    }
    float sum = 0.0f;
    #pragma unroll
    for (int k = 0; k < K_; ++k) { t[k] = __expf(t[k] - mx); sum += t[k]; }
    const float inv = 1.0f / sum;
    float* outA = wsA + ((size_t)b * N_ + n) * K_;
    #pragma unroll
    for (int k = 0; k < K_; ++k) {
      const float av = t[k] * inv;
      outA[k] = av;
      atomicAdd(&sAsum[k], av);              // ds_add_f32
    }
  }
  __syncthreads();
  if (tid < K_) atomicAdd(&wsAsum[b * K_ + tid], sAsum[tid]);
}

// -------------------------------------------------------------------------
// Pass 2: E[b,k,d] = sum_n A[b,n,k]*Xf[b,n,d] - asum[b,k]*C[k,d]
//   Block = (batch, 16-wide d-tile). 8 waves: 2 k-tiles x 4 n-quarters,
//   each wave runs 256 WMMA f32 16x16x4 ops, LDS cross-wave reduction.
// -------------------------------------------------------------------------
__global__ __launch_bounds__(256) void enc_aggregate_kernel(
    const float* __restrict__ X,       // (B, D, N)
    const float* __restrict__ Cw,      // (K, D)
    const float* __restrict__ wsA,     // (B, N, K)
    const float* __restrict__ wsAsum,  // (B, K)
    float* __restrict__ Eout)          // (B, K, D)
{
  __shared__ float sAcc[8][16 * 16];   // 8 KB partials

  const int b  = blockIdx.x >> 4;      // 16 d-tiles per batch
  const int dt = blockIdx.x & 15;
  const int d0 = dt * 16;

  const int tid   = threadIdx.x;
  const int wave  = tid >> 5;
  const int lane  = tid & 31;
  const int m     = lane & 15;
  const int lh    = lane >> 4;
  const int ktile = wave >> 2;         // 0..1
  const int q     = wave & 3;          // n quarter
  const int k0    = ktile * 16;

  const float* Ab = wsA + (size_t)b * N_ * K_;
  const float* Xb = X + (size_t)b * D_ * N_ + (size_t)d0 * N_;

  v8f acc = {};
  const int nEnd = q * 1024 + 1024;
  for (int nn = q * 1024; nn < nEnd; nn += 4) {
    const int r = nn + 2 * lh;
    // A frag: rows = 16 codewords, cols = 4 n's. (k0+m, n=r / r+1)
    v2f a;
    a.x = Ab[(size_t)r * K_ + k0 + m];
    a.y = Ab[(size_t)(r + 1) * K_ + k0 + m];
    // B frag: rows n=r,r+1 (contiguous in memory -> global_load_b64), col d0+m
    v2f bb = *(const v2f*)&Xb[(size_t)m * N_ + r];
    acc = __builtin_amdgcn_wmma_f32_16x16x4_f32(false, a, false, bb, (short)0, acc, false, false);
  }

  #pragma unroll
  for (int v = 0; v < 8; ++v)
    sAcc[wave][(v + 8 * lh) * 16 + m] = acc[v];
  __syncthreads();

  // 512 outputs (32 k x 16 d), fuse "- asum * C" on store
  for (int idx = tid; idx < K_ * 16; idx += 256) {
    const int k  = idx >> 4;
    const int dd = idx & 15;
    const int t  = k >> 4;               // which k-tile
    const int kr = k & 15;
    float s = 0.0f;
    #pragma unroll
    for (int qq = 0; qq < 4; ++qq) s += sAcc[t * 4 + qq][kr * 16 + dd];
    const float e = s - wsAsum[b * K_ + k] * Cw[k * D_ + d0 + dd];
    Eout[((size_t)b * K_ + k) * D_ + d0 + dd] = e;
  }
}

// -------------------------------------------------------------------------
extern "C" void kernel_launch(void* const* d_in, const int* in_sizes, int n_in,
                              void* d_out, int out_size, void* d_ws, size_t ws_size,
                              hipStream_t stream) {
  const float* X  = (const float*)d_in[0];   // (B, D, H, W) fp32
  const float* Cw = (const float*)d_in[1];   // (K, D) fp32
  const float* Sc = (const float*)d_in[2];   // (K,) fp32
  float* Eout = (float*)d_out;               // (B, K, D) fp32

  float* wsA    = (float*)d_ws;                      // B*N*K floats = 2 MB
  float* wsAsum = wsA + (size_t)B_ * N_ * K_;        // B*K floats

  enc_zero_kernel<<<1, 128, 0, stream>>>(wsAsum, B_ * K_);
  enc_assign_kernel<<<B_ * (N_ / 128), 256, 0, stream>>>(X, Cw, Sc, wsA, wsAsum);
  enc_aggregate_kernel<<<B_ * (D_ / 16), 256, 0, stream>>>(X, Cw, wsA, wsAsum, Eout);
}